// mGCNLayer_14044543058525
// MI455X (gfx1250) — compile-verified
//
#include <hip/hip_runtime.h>
#include <hip/hip_bf16.h>

// Problem constants (from reference)
#define Nn 20000
#define Ee 640000
#define Hh 256
static constexpr size_t NH = (size_t)Nn * Hh;   // 5,120,000 floats per modality

typedef __attribute__((ext_vector_type(2))) float v2f;
typedef __attribute__((ext_vector_type(8))) float v8f;

// ---------------------------------------------------------------------------
// Utility: zero a float buffer (graph-capture safe, no memset API needed)
// ---------------------------------------------------------------------------
__global__ __launch_bounds__(256) void zero_kernel(float* __restrict__ p, int n) {
    int i = blockIdx.x * 256 + threadIdx.x;
    if (i < n) p[i] = 0.0f;
}

// ---------------------------------------------------------------------------
// Degree computation: deg_out[src]++ , deg_in[dst]++
// ---------------------------------------------------------------------------
__global__ __launch_bounds__(256) void degree_kernel(const int* __restrict__ src,
                                                     const int* __restrict__ dst,
                                                     float* deg_out, float* deg_in) {
    int e = blockIdx.x * 256 + threadIdx.x;
    if (e < Ee) {
        atomicAdd(&deg_out[src[e]], 1.0f);
        atomicAdd(&deg_in[dst[e]], 1.0f);
    }
}

// dinv = rsqrt(max(deg, 1))  (covers both deg_out and deg_in in one launch)
__global__ __launch_bounds__(256) void dinv_kernel(const float* __restrict__ deg,
                                                   float* __restrict__ dinv, int n) {
    int i = blockIdx.x * 256 + threadIdx.x;
    if (i < n) dinv[i] = rsqrtf(fmaxf(deg[i], 1.0f));
}

// ---------------------------------------------------------------------------
// WMMA f32 GEMM:  C[N,256] = epilogue( rowscale ⊙ (A[N,K] @ B[K,256]) + bias )
//
// One wave -> 16x64 output tile (4 accumulators, A-fragment reused 4x).
// V_WMMA_F32_16X16X4_F32 layout (wave32):
//   A 16x4 : lane m=lane&15; kh = (lane>=16)?2:0 ; a = {A[m][k+kh], A[m][k+kh+1]}
//   B 4x16 : lane n=lane&15; b = {B[k+kh][n],      B[k+kh+1][n]}
//   C 16x16: VGPR r, lanes0-15 -> M=r, lanes16-31 -> M=r+8, N=lane&15
// Flags:
//   RELU    : relu epilogue
//   COMBINE : C = 0.5*(gemm+bias) + 0.5*addin        (final alpha-blend)
//   RSCALE  : multiply row r of (A@B) by rowscale[r] (GCN D_in^{-1/2}, commutes)
//   BLEND   : A_eff = w0*A + w1*A(+NH) + w2*A(+2NH)  (cross-modal einsum fused)
// ---------------------------------------------------------------------------
template <bool RELU, bool COMBINE, bool RSCALE, bool BLEND>
__global__ __launch_bounds__(256)
void gemm_kernel(const float* __restrict__ A, int K,
                 const float* __restrict__ B,
                 const float* __restrict__ bias,
                 const float* __restrict__ rowscale,
                 const float* addin,
                 const float* __restrict__ wrow,
                 float* C) {
    const int lane  = threadIdx.x & 31;
    const int wave  = threadIdx.x >> 5;
    const int wtile = blockIdx.x * 8 + wave;     // 0 .. 4999
    const int row0  = (wtile >> 2) * 16;         // 1250 row tiles
    const int col0  = (wtile & 3) * 64;          // 4 col blocks of 64
    const int m     = lane & 15;
    const int kh    = (lane >> 4) * 2;           // 0 or 2

    float w0 = 0.f, w1 = 0.f, w2 = 0.f;
    if (BLEND) { w0 = wrow[0]; w1 = wrow[1]; w2 = wrow[2]; }

    const float* Ar0 = A + (size_t)(row0 + m) * K;
    const float* Ar1 = BLEND ? (A + NH + (size_t)(row0 + m) * K) : nullptr;
    const float* Ar2 = BLEND ? (A + 2 * NH + (size_t)(row0 + m) * K) : nullptr;

    v8f acc0 = {}, acc1 = {}, acc2 = {}, acc3 = {};

    for (int k = 0; k < K; k += 4) {
        v2f a;
        if (BLEND) {
            a.x = w0 * Ar0[k + kh]     + w1 * Ar1[k + kh]     + w2 * Ar2[k + kh];
            a.y = w0 * Ar0[k + kh + 1] + w1 * Ar1[k + kh + 1] + w2 * Ar2[k + kh + 1];
        } else {
            a.x = Ar0[k + kh];
            a.y = Ar0[k + kh + 1];
        }
        const float* B0 = B + (size_t)(k + kh) * Hh + col0 + m;
        const float* B1 = B0 + Hh;
        v2f b;
        b.x = B0[0];  b.y = B1[0];
        acc0 = __builtin_amdgcn_wmma_f32_16x16x4_f32(false, a, false, b, (short)0, acc0, false, false);
        b.x = B0[16]; b.y = B1[16];
        acc1 = __builtin_amdgcn_wmma_f32_16x16x4_f32(false, a, false, b, (short)0, acc1, false, false);
        b.x = B0[32]; b.y = B1[32];
        acc2 = __builtin_amdgcn_wmma_f32_16x16x4_f32(false, a, false, b, (short)0, acc2, false, false);
        b.x = B0[48]; b.y = B1[48];
        acc3 = __builtin_amdgcn_wmma_f32_16x16x4_f32(false, a, false, b, (short)0, acc3, false, false);
    }

    const int mh = (lane >> 4) * 8;
#pragma unroll
    for (int s4 = 0; s4 < 4; ++s4) {
        v8f acc = (s4 == 0) ? acc0 : (s4 == 1) ? acc1 : (s4 == 2) ? acc2 : acc3;
        const int col = col0 + s4 * 16 + m;
        const float bv = bias ? bias[col] : 0.0f;
#pragma unroll
        for (int r = 0; r < 8; ++r) {
            const int row = row0 + r + mh;
            float v = acc[r];
            if (RSCALE) v *= rowscale[row];
            v += bv;
            if (RELU) v = fmaxf(v, 0.0f);
            const size_t idx = (size_t)row * Hh + col;
            if (COMBINE) v = 0.5f * v + 0.5f * addin[idx];
            C[idx] = v;
        }
    }
}

// ---------------------------------------------------------------------------
// Edge scatter: agg_m[dst] += proj_m[src] * doutinv[src], all 3 modalities.
// 64 lanes / edge, float4 per lane. proj (61MB) lives in the 192MB L2 ->
// random gathers are L2-resident, not HBM bound.
// ---------------------------------------------------------------------------
__global__ __launch_bounds__(256)
void scatter_kernel(const int* __restrict__ src, const int* __restrict__ dst,
                    const float* __restrict__ proj, const float* __restrict__ doutinv,
                    float* agg) {
    const int e = blockIdx.x * 4 + (threadIdx.x >> 6);
    const int t = (threadIdx.x & 63) << 2;
    const int s = src[e];
    const int d = dst[e];
    const float sc = doutinv[s];
#pragma unroll
    for (int mm = 0; mm < 3; ++mm) {
        const float4 v = *(const float4*)(proj + ((size_t)mm * Nn + s) * Hh + t);
        float* ap = agg + ((size_t)mm * Nn + d) * Hh + t;
        atomicAdd(ap + 0, v.x * sc);
        atomicAdd(ap + 1, v.y * sc);
        atomicAdd(ap + 2, v.z * sc);
        atomicAdd(ap + 3, v.w * sc);
    }
}

// ---------------------------------------------------------------------------
// Cosine-similarity score accumulation: one wave32 per node, shfl reduction,
// 9 float atomics per node (scores[d][e] += dot_de/(|d||e|) / N).
// ---------------------------------------------------------------------------
__global__ __launch_bounds__(256)
void score_kernel(const float* __restrict__ T, float* __restrict__ scores) {
    const int lane = threadIdx.x & 31;
    const int node = blockIdx.x * 8 + (threadIdx.x >> 5);
    const float* t0 = T + (size_t)node * Hh;
    const float* t1 = t0 + NH;
    const float* t2 = t1 + NH;
    float d00 = 0, d11 = 0, d22 = 0, d01 = 0, d02 = 0, d12 = 0;
    const int base = lane * 8;
#pragma unroll
    for (int j = 0; j < 8; ++j) {
        const float a = t0[base + j], b = t1[base + j], c = t2[base + j];
        d00 = fmaf(a, a, d00); d11 = fmaf(b, b, d11); d22 = fmaf(c, c, d22);
        d01 = fmaf(a, b, d01); d02 = fmaf(a, c, d02); d12 = fmaf(b, c, d12);
    }
#pragma unroll
    for (int off = 16; off > 0; off >>= 1) {
        d00 += __shfl_xor(d00, off, 32);
        d11 += __shfl_xor(d11, off, 32);
        d22 += __shfl_xor(d22, off, 32);
        d01 += __shfl_xor(d01, off, 32);
        d02 += __shfl_xor(d02, off, 32);
        d12 += __shfl_xor(d12, off, 32);
    }
    if (lane == 0) {
        const float EPS = 1e-12f, invN = 1.0f / (float)Nn;
        const float n0 = fmaxf(sqrtf(d00), EPS);
        const float n1 = fmaxf(sqrtf(d11), EPS);
        const float n2 = fmaxf(sqrtf(d22), EPS);
        const float s00 = d00 / (n0 * n0), s11 = d11 / (n1 * n1), s22 = d22 / (n2 * n2);
        const float s01 = d01 / (n0 * n1), s02 = d02 / (n0 * n2), s12 = d12 / (n1 * n2);
        atomicAdd(&scores[0], s00 * invN); atomicAdd(&scores[1], s01 * invN); atomicAdd(&scores[2], s02 * invN);
        atomicAdd(&scores[3], s01 * invN); atomicAdd(&scores[4], s11 * invN); atomicAdd(&scores[5], s12 * invN);
        atomicAdd(&scores[6], s02 * invN); atomicAdd(&scores[7], s12 * invN); atomicAdd(&scores[8], s22 * invN);
    }
}

// 3x3 row softmax (tiny)
__global__ void softmax_kernel(const float* __restrict__ scores, float* __restrict__ w) {
    if (threadIdx.x == 0 && blockIdx.x == 0) {
        for (int d = 0; d < 3; ++d) {
            const float a = scores[d * 3], b = scores[d * 3 + 1], c = scores[d * 3 + 2];
            const float mx = fmaxf(a, fmaxf(b, c));
            const float ea = expf(a - mx), eb = expf(b - mx), ec = expf(c - mx);
            const float s = ea + eb + ec;
            w[d * 3] = ea / s; w[d * 3 + 1] = eb / s; w[d * 3 + 2] = ec / s;
        }
    }
}

// ---------------------------------------------------------------------------
// Orchestration
// Input dict order: per modality i: f,Wp,bp,Wg,bg,Wa ; then Wl,bl,src,dst
// ---------------------------------------------------------------------------
extern "C" void kernel_launch(void* const* d_in, const int* in_sizes, int n_in,
                              void* d_out, int out_size, void* d_ws, size_t ws_size,
                              hipStream_t stream) {
    (void)in_sizes; (void)n_in; (void)out_size; (void)ws_size;

    const float* f[3]  = {(const float*)d_in[0], (const float*)d_in[6],  (const float*)d_in[12]};
    const float* Wp[3] = {(const float*)d_in[1], (const float*)d_in[7],  (const float*)d_in[13]};
    const float* bp[3] = {(const float*)d_in[2], (const float*)d_in[8],  (const float*)d_in[14]};
    const float* Wg[3] = {(const float*)d_in[3], (const float*)d_in[9],  (const float*)d_in[15]};
    const float* bg[3] = {(const float*)d_in[4], (const float*)d_in[10], (const float*)d_in[16]};
    const float* Wa[3] = {(const float*)d_in[5], (const float*)d_in[11], (const float*)d_in[17]};
    const float* Wl = (const float*)d_in[18];
    const float* bl = (const float*)d_in[19];
    const int* src  = (const int*)d_in[20];
    const int* dst  = (const int*)d_in[21];
    const int Kd[3] = {512, 256, 128};

    float* ws     = (float*)d_ws;
    float* proj   = ws;                     // 3*NH : relu(f @ Wp + bp)
    float* buf    = ws + 3 * NH;            // 3*NH : T, then reused as agg
    float* deg    = ws + 6 * NH;            // 2*N  : deg_out, deg_in
    float* dinv   = deg + 2 * Nn;           // 2*N  : doutinv, dininv
    float* scores = dinv + 2 * Nn;          // 16
    float* wmat   = scores + 16;            // 16
    float* out    = (float*)d_out;          // 3*NH : intra staged here, then blended

    const int GB = 625;  // 5000 wave-tiles / 8 waves per block

    // degrees + normalization factors
    zero_kernel<<<(2 * Nn + 255) / 256, 256, 0, stream>>>(deg, 2 * Nn);
    zero_kernel<<<1, 256, 0, stream>>>(scores, 32);
    degree_kernel<<<(Ee + 255) / 256, 256, 0, stream>>>(src, dst, deg, deg + Nn);
    dinv_kernel<<<(2 * Nn + 255) / 256, 256, 0, stream>>>(deg, dinv, 2 * Nn);

    // proj_m = relu(f_m @ Wp_m + bp_m)
    for (int m = 0; m < 3; ++m)
        gemm_kernel<true, false, false, false><<<GB, 256, 0, stream>>>(
            f[m], Kd[m], Wp[m], bp[m], nullptr, nullptr, nullptr, proj + (size_t)m * NH);

    // T_m = proj_m @ Wa_m  -> scores -> softmax weights
    for (int m = 0; m < 3; ++m)
        gemm_kernel<false, false, false, false><<<GB, 256, 0, stream>>>(
            proj + (size_t)m * NH, Hh, Wa[m], nullptr, nullptr, nullptr, nullptr, buf + (size_t)m * NH);
    score_kernel<<<Nn / 8, 256, 0, stream>>>(buf, scores);
    softmax_kernel<<<1, 32, 0, stream>>>(scores, wmat);

    // agg_m[dst] += (proj_m * doutinv)[src]   (buf reused as agg)
    zero_kernel<<<(int)((3 * NH + 255) / 256), 256, 0, stream>>>(buf, (int)(3 * NH));
    scatter_kernel<<<Ee / 4, 256, 0, stream>>>(src, dst, proj, dinv, buf);

    // intra_m = relu(dininv ⊙ (agg_m @ Wg_m) + bg_m)  -> staged in d_out
    for (int m = 0; m < 3; ++m)
        gemm_kernel<true, false, true, false><<<GB, 256, 0, stream>>>(
            buf + (size_t)m * NH, Hh, Wg[m], bg[m], dinv + Nn, nullptr, nullptr, out + (size_t)m * NH);

    // out_d = 0.5*intra_d + 0.5*((Σ_g w[d,g] proj_g) @ Wl + bl)
    for (int d = 0; d < 3; ++d)
        gemm_kernel<false, true, false, true><<<GB, 256, 0, stream>>>(
            proj, Hh, Wl, bl, nullptr, out + (size_t)d * NH, wmat + 3 * d, out + (size_t)d * NH);
}